// Convolve_4509715661235
// MI455X (gfx1250) — compile-verified
//
#include <hip/hip_runtime.h>

typedef __attribute__((ext_vector_type(2))) float v2f;
typedef __attribute__((ext_vector_type(4))) float v4f;
typedef __attribute__((ext_vector_type(8))) float v8f;

#define ALPHA 0.3f
#define EPSF  1e-6f

// Problem dims (hardcoded from reference setup_inputs).
#define BB   2
#define NN   8192
#define KK   32
#define CC   128
#define HH   128
#define MM   (BB * NN)   // 16384 rows

__device__ __forceinline__ float lrelu(float x) { return x >= 0.f ? x : ALPHA * x; }

// ---------------------------------------------------------------------------
// Kernel 1: hidden[m, :] = leaky_relu(emb[m, :] @ Qw + Qb)   (M=16384, K=C=128, N=H=128)
// Wave-per-16x16-tile, 8 waves per block cover 128 rows x 16 cols.
// f32 WMMA: D(16x16) += A(16x4) * B(4x16), 32 k-steps.
// ---------------------------------------------------------------------------
__global__ __launch_bounds__(256) void k1_hidden(const float* __restrict__ emb,
                                                 const float* __restrict__ Qw,
                                                 const float* __restrict__ Qb,
                                                 float* __restrict__ hidden)
{
    const int lane = threadIdx.x & 31;
    const int wave = threadIdx.x >> 5;
    const int m0   = (blockIdx.x * 8 + wave) << 4;   // 16-row tile
    const int h0   = blockIdx.y << 4;                // 16-col tile
    const int rc   = lane & 15;                      // A row / B col / C col
    const int half = lane >> 4;

    const float* arow = emb + (size_t)(m0 + rc) * CC;
    v8f acc = (v8f)0.f;

    #pragma unroll 4
    for (int k4 = 0; k4 < CC; k4 += 4) {
        const int k0 = k4 + 2 * half;                // this lane's K pair
        v2f a = *(const v2f*)(arow + k0);            // A[rc, k0..k0+1] (b64 load)
        v2f b;
        b.x = Qw[(size_t)k0 * HH + h0 + rc];         // B[k0,   col]
        b.y = Qw[(size_t)(k0 + 1) * HH + h0 + rc];   // B[k0+1, col]
        acc = __builtin_amdgcn_wmma_f32_16x16x4_f32(false, a, false, b,
                                                    (short)0, acc, false, false);
    }

    const float bv = Qb[h0 + rc];
    #pragma unroll
    for (int r = 0; r < 8; ++r) {
        const int m = m0 + r + (half << 3);          // C/D layout: row = r + 8*half
        hidden[(size_t)m * HH + h0 + rc] = lrelu(acc[r] + bv);
    }
}

// ---------------------------------------------------------------------------
// Kernel 2: wsh[b,n,:] = (sum_k w[n,nbr]*hidden[b,nbr,:]) / (sum_k w + eps)
// One wave per (b,n); lane k loads neighbor k's id+weight, broadcast by shfl;
// each lane accumulates 4 contiguous H channels (float4 gathers, L2-resident).
// ---------------------------------------------------------------------------
__global__ __launch_bounds__(256) void k2_gather(const float* __restrict__ hidden,
                                                 const float* __restrict__ weights,
                                                 const int* __restrict__ nbr,
                                                 float* __restrict__ wsh)
{
    const int lane = threadIdx.x & 31;
    const int wave = threadIdx.x >> 5;
    const int idx  = blockIdx.x * 8 + wave;          // 0 .. B*N-1
    const int b    = idx >> 13;                      // / N (N = 8192)
    const int n    = idx & (NN - 1);

    const int   j = nbr[(size_t)n * KK + lane];      // my neighbor id
    const float w = weights[(size_t)n * NN + j];     // my neighbor weight

    const float* hb = hidden + (size_t)b * NN * HH + (size_t)lane * 4;

    v4f   acc  = (v4f)0.f;
    float wsum = 0.f;
    #pragma unroll 8
    for (int k = 0; k < KK; ++k) {
        const int   jk = __shfl(j, k, 32);
        const float wk = __shfl(w, k, 32);
        const v4f h = *(const v4f*)(hb + (size_t)jk * HH);
        acc.x += wk * h.x;
        acc.y += wk * h.y;
        acc.z += wk * h.z;
        acc.w += wk * h.w;
        wsum  += wk;
    }
    const float inv = 1.f / (wsum + EPSF);
    v4f r;
    r.x = acc.x * inv; r.y = acc.y * inv; r.z = acc.z * inv; r.w = acc.w * inv;
    *(v4f*)(wsh + (size_t)idx * HH + lane * 4) = r;
}

// ---------------------------------------------------------------------------
// Kernel 3: out[m,:] = leaky_relu(emb[m,:] @ Ww[0:128,:] + wsh[m,:] @ Ww[128:256,:] + Wb)
// Concat is folded: two 128-deep WMMA reduction loops into one accumulator.
// Pre-norm result written straight to d_out (normalized in place by k4).
// ---------------------------------------------------------------------------
__global__ __launch_bounds__(256) void k3_gemm2(const float* __restrict__ emb,
                                                const float* __restrict__ wsh,
                                                const float* __restrict__ Ww,
                                                const float* __restrict__ Wb,
                                                float* __restrict__ out)
{
    const int lane = threadIdx.x & 31;
    const int wave = threadIdx.x >> 5;
    const int m0   = (blockIdx.x * 8 + wave) << 4;
    const int h0   = blockIdx.y << 4;
    const int rc   = lane & 15;
    const int half = lane >> 4;

    const float* a1 = emb + (size_t)(m0 + rc) * CC;
    const float* a2 = wsh + (size_t)(m0 + rc) * HH;
    v8f acc = (v8f)0.f;

    #pragma unroll 4
    for (int k4 = 0; k4 < CC; k4 += 4) {
        const int k0 = k4 + 2 * half;
        v2f a = *(const v2f*)(a1 + k0);
        v2f b;
        b.x = Ww[(size_t)k0 * HH + h0 + rc];
        b.y = Ww[(size_t)(k0 + 1) * HH + h0 + rc];
        acc = __builtin_amdgcn_wmma_f32_16x16x4_f32(false, a, false, b,
                                                    (short)0, acc, false, false);
    }
    #pragma unroll 4
    for (int k4 = 0; k4 < HH; k4 += 4) {
        const int k0 = k4 + 2 * half;
        v2f a = *(const v2f*)(a2 + k0);
        v2f b;
        b.x = Ww[(size_t)(CC + k0) * HH + h0 + rc];
        b.y = Ww[(size_t)(CC + k0 + 1) * HH + h0 + rc];
        acc = __builtin_amdgcn_wmma_f32_16x16x4_f32(false, a, false, b,
                                                    (short)0, acc, false, false);
    }

    const float bv = Wb[h0 + rc];
    #pragma unroll
    for (int r = 0; r < 8; ++r) {
        const int m = m0 + r + (half << 3);
        out[(size_t)m * HH + h0 + rc] = lrelu(acc[r] + bv);
    }
}

// ---------------------------------------------------------------------------
// Kernel 4: out[m,:] /= (||out[m,:]||_2 + eps)   (in place; one wave per row)
// ---------------------------------------------------------------------------
__global__ __launch_bounds__(256) void k4_norm(float* __restrict__ out)
{
    const int lane = threadIdx.x & 31;
    const int wave = threadIdx.x >> 5;
    const int row  = blockIdx.x * 8 + wave;

    float* p = out + (size_t)row * HH + lane * 4;
    v4f v = *(const v4f*)p;
    float s = v.x * v.x + v.y * v.y + v.z * v.z + v.w * v.w;
    #pragma unroll
    for (int off = 16; off >= 1; off >>= 1)
        s += __shfl_xor(s, off, 32);
    const float inv = 1.f / (sqrtf(s) + EPSF);
    v.x *= inv; v.y *= inv; v.z *= inv; v.w *= inv;
    *(v4f*)p = v;
}

// ---------------------------------------------------------------------------
extern "C" void kernel_launch(void* const* d_in, const int* in_sizes, int n_in,
                              void* d_out, int out_size, void* d_ws, size_t ws_size,
                              hipStream_t stream)
{
    const float* emb     = (const float*)d_in[0];   // (B,N,C)
    const float* weights = (const float*)d_in[1];   // (N,N)
    const int*   nbr     = (const int*)  d_in[2];   // (N,K)
    const float* Qw      = (const float*)d_in[3];   // (C,H)
    const float* Qb      = (const float*)d_in[4];   // (H)
    const float* Ww      = (const float*)d_in[5];   // (C+H,H)
    const float* Wb      = (const float*)d_in[6];   // (H)
    float*       out     = (float*)d_out;           // (B,N,H)

    float* hidden = (float*)d_ws;                   // 16384*128 f32 = 8 MB
    float* wsh    = hidden + (size_t)MM * HH;       // next 8 MB

    const dim3 blk(256);
    // GEMM grids: x = M/(16 rows * 8 waves) = 128 blocks, y = H/16 = 8 tiles.
    k1_hidden<<<dim3(MM / 128, HH / 16), blk, 0, stream>>>(emb, Qw, Qb, hidden);
    // One wave per (b,n): B*N/8 blocks.
    k2_gather<<<dim3(MM / 8), blk, 0, stream>>>(hidden, weights, nbr, wsh);
    k3_gemm2<<<dim3(MM / 128, HH / 16), blk, 0, stream>>>(emb, wsh, Ww, Wb, out);
    k4_norm<<<dim3(MM / 8), blk, 0, stream>>>(out);
}